// MaskCreatorForward_53807350284480
// MI455X (gfx1250) — compile-verified
//
#include <hip/hip_runtime.h>
#include <hip/hip_bf16.h>

// ---------------------------------------------------------------------------
// MaskCreatorForward for MI455X (gfx1250, wave32)
//   k1: 20-iteration fixed-point flow inversion + |f|+|g|   (L2-resident gathers)
//   k2: horizontal 19-tap Gaussian blur via v_wmma_f32_16x16x4_f32
//   k3: vertical   19-tap Gaussian blur via v_wmma_f32_16x16x4_f32 + threshold
// ---------------------------------------------------------------------------

typedef __attribute__((ext_vector_type(2))) float v2f;
typedef __attribute__((ext_vector_type(8))) float v8f;

#define IMG_W 512
#define IMG_H 512
#define IMG_HW (IMG_W * IMG_H)
#define NBATCH 16
#define NITERS 20

// Normalized 19-tap Gaussian, sigma=3 (precomputed, sum == 1)
__device__ const float GW[19] = {
    0.0014794f, 0.0038041f, 0.0087534f, 0.0180237f, 0.0332077f,
    0.0547503f, 0.0807754f, 0.1066390f, 0.1259791f, 0.1331760f,
    0.1259791f, 0.1066390f, 0.0807754f, 0.0547503f, 0.0332077f,
    0.0180237f, 0.0087534f, 0.0038041f, 0.0014794f};

__device__ __forceinline__ float gtap(int d) {
    // g[d] for 0<=d<=18, else 0 (branchless: clamp + select)
    unsigned ud = (unsigned)d;
    float w = GW[ud <= 18u ? ud : 0u];
    return (ud <= 18u) ? w : 0.0f;
}

__device__ __forceinline__ int reflect512(int i) {
    i = (i < 0) ? -i : i;                 // reflect (no edge repeat): -1 -> 1
    i = (i > 511) ? (1022 - i) : i;       // 512 -> 510
    return i;
}

// --------------------------- Kernel 1: inversion ---------------------------
// One thread per pixel; g kept in registers; 20 dependent bilinear gathers.
// Flow plane is 2 MB/batch -> L2-resident on MI455X (192 MB L2).
__global__ void __launch_bounds__(256) invert_mag_kernel(
        const float* __restrict__ flow, float* __restrict__ summed) {
    int idx = blockIdx.x * blockDim.x + threadIdx.x;   // 16*512*512 threads
    int b = idx >> 18;
    int p = idx & (IMG_HW - 1);
    int y = p >> 9;
    int x = p & 511;

    const float* fx = flow + (size_t)b * 2 * IMG_HW;   // channel 0 (x-disp)
    const float* fy = fx + IMG_HW;                     // channel 1 (y-disp)

    float gx = 0.0f, gy = 0.0f;
    #pragma unroll 1
    for (int it = 0; it < NITERS; ++it) {
        float px = fminf(fmaxf((float)x + gx, 0.0f), 511.0f);
        float py = fminf(fmaxf((float)y + gy, 0.0f), 511.0f);
        float fx0 = floorf(px), fy0 = floorf(py);
        int x0 = (int)fx0, y0 = (int)fy0;
        int x1 = min(x0 + 1, 511), y1 = min(y0 + 1, 511);
        float wx = px - fx0, wy = py - fy0;
        int i00 = (y0 << 9) + x0;
        int i01 = (y0 << 9) + x1;
        int i10 = (y1 << 9) + x0;
        int i11 = (y1 << 9) + x1;
        float ax = fx[i00], bx = fx[i01], cx = fx[i10], dx = fx[i11];
        float ay = fy[i00], by = fy[i01], cy = fy[i10], dy = fy[i11];
        float tx = ax + (bx - ax) * wx;
        float ux = cx + (dx - cx) * wx;
        float sx = tx + (ux - tx) * wy;
        float ty = ay + (by - ay) * wx;
        float uy = cy + (dy - cy) * wx;
        float sy = ty + (uy - ty) * wy;
        gx = -sx;
        gy = -sy;
    }
    float ffx = fx[p], ffy = fy[p];
    float mF = sqrtf(ffx * ffx + ffy * ffy);
    float mB = sqrtf(gx * gx + gy * gy);
    summed[(size_t)b * IMG_HW + p] = mF + mB;
}

// ------------------- Kernel 2: horizontal blur (WMMA) ----------------------
// Per wave: one 16x16 tile.  D(16x16) = A(16x36) x G(36x16), 9 K=4 WMMA steps.
// A fragment (32-bit 16x4 layout): lane = m + 16*(k>=2); v0 holds K=base,
// v1 holds K=base+1 with base = 2*(lane>>4).  G rows 34,35 are zero (tap>18).
__global__ void __launch_bounds__(256) hblur_wmma_kernel(
        const float* __restrict__ src, float* __restrict__ dst) {
    int wave = blockIdx.x * (blockDim.x >> 5) + (threadIdx.x >> 5);
    int lane = threadIdx.x & 31;
    int half = lane >> 4;
    int lm = lane & 15;

    int tc = wave & 31;          // 32 tile-cols
    int tr = (wave >> 5) & 31;   // 32 tile-rows
    int b = wave >> 10;          // 16 batches
    int r0 = tr << 4, c0 = tc << 4;

    const float* rowp = src + (size_t)b * IMG_HW + (r0 + lm) * IMG_W;

    v8f acc = {};
    #pragma unroll
    for (int kk = 0; kk < 9; ++kk) {
        int k0 = kk * 4 + half * 2;          // global K index of v0 element
        // A: in_ext[lm, k0], in_ext[lm, k0+1]   (reflect-padded columns)
        int j0 = reflect512(c0 - 9 + k0);
        int j1 = reflect512(c0 - 9 + k0 + 1);
        v2f a;
        a.x = rowp[j0];
        a.y = rowp[j1];
        // B: G[k0, lm], G[k0+1, lm] with G[j,c] = g[j-c]
        v2f bm;
        bm.x = gtap(k0 - lm);
        bm.y = gtap(k0 + 1 - lm);
        acc = __builtin_amdgcn_wmma_f32_16x16x4_f32(
            false, a, false, bm, (short)0, acc, false, false);
    }

    // D layout: VGPR i -> row i (lanes 0-15) / row i+8 (lanes 16-31), col = lm
    float* ocol = dst + (size_t)b * IMG_HW + c0 + lm;
    #pragma unroll
    for (int i = 0; i < 8; ++i) {
        int rr = r0 + i + half * 8;
        ocol[rr * IMG_W] = acc[i];
    }
}

// -------------- Kernel 3: vertical blur (WMMA) + threshold -----------------
// D(16x16) = Gv(16x36) x in_ext(36x16),  Gv[r,j] = g[j-r].
__global__ void __launch_bounds__(256) vblur_mask_wmma_kernel(
        const float* __restrict__ src, float* __restrict__ dst) {
    int wave = blockIdx.x * (blockDim.x >> 5) + (threadIdx.x >> 5);
    int lane = threadIdx.x & 31;
    int half = lane >> 4;
    int lm = lane & 15;

    int tc = wave & 31;
    int tr = (wave >> 5) & 31;
    int b = wave >> 10;
    int r0 = tr << 4, c0 = tc << 4;

    const float* plane = src + (size_t)b * IMG_HW;

    v8f acc = {};
    #pragma unroll
    for (int kk = 0; kk < 9; ++kk) {
        int k0 = kk * 4 + half * 2;
        // A: Gv[lm, k0], Gv[lm, k0+1] = g[k0-lm], g[k0+1-lm]
        v2f a;
        a.x = gtap(k0 - lm);
        a.y = gtap(k0 + 1 - lm);
        // B: in_ext[k0, lm], in_ext[k0+1, lm]   (reflect-padded rows)
        int rA = reflect512(r0 - 9 + k0);
        int rB = reflect512(r0 - 9 + k0 + 1);
        v2f bm;
        bm.x = plane[rA * IMG_W + c0 + lm];
        bm.y = plane[rB * IMG_W + c0 + lm];
        acc = __builtin_amdgcn_wmma_f32_16x16x4_f32(
            false, a, false, bm, (short)0, acc, false, false);
    }

    float* ocol = dst + (size_t)b * IMG_HW + c0 + lm;
    #pragma unroll
    for (int i = 0; i < 8; ++i) {
        int rr = r0 + i + half * 8;
        ocol[rr * IMG_W] = (acc[i] >= 1.0f) ? 1.0f : 0.0f;
    }
}

// ---------------------------------------------------------------------------
extern "C" void kernel_launch(void* const* d_in, const int* in_sizes, int n_in,
                              void* d_out, int out_size, void* d_ws, size_t ws_size,
                              hipStream_t stream) {
    (void)in_sizes; (void)n_in; (void)out_size; (void)ws_size;
    const float* flow = (const float*)d_in[0];
    float* out = (float*)d_out;

    float* ws_summed = (float*)d_ws;                       // [16,512,512]
    float* ws_hblur  = ws_summed + (size_t)NBATCH * IMG_HW; // [16,512,512]

    // Kernel 1: 16*512*512 pixels, one thread each
    int total = NBATCH * IMG_HW;
    invert_mag_kernel<<<total / 256, 256, 0, stream>>>(flow, ws_summed);

    // Blur passes: 16*32*32 = 16384 tiles, 1 wave per tile, 8 waves per block
    int tiles = NBATCH * 32 * 32;
    hblur_wmma_kernel<<<tiles / 8, 256, 0, stream>>>(ws_summed, ws_hblur);
    vblur_mask_wmma_kernel<<<tiles / 8, 256, 0, stream>>>(ws_hblur, out);
}